// kp_3dgs_loss_7249904795878
// MI455X (gfx1250) — compile-verified
//
#include <hip/hip_runtime.h>
#include <hip/hip_bf16.h>
#include <math.h>

// Problem constants from the reference
#define BB   8
#define NN   256
#define SS   128
#define MM   8192
#define KK   200
#define KP   208          // KK padded to a multiple of 16 for WMMA tiles
#define EPSF 1e-9f

typedef __attribute__((ext_vector_type(2))) float v2f;
typedef __attribute__((ext_vector_type(8))) float v8f;

// Monotone float <-> uint mapping (total order on floats incl. negatives)
__device__ __forceinline__ unsigned f2ord(float f) {
  unsigned b = __float_as_uint(f);
  return (b & 0x80000000u) ? ~b : (b | 0x80000000u);
}
__device__ __forceinline__ float ord2f(unsigned u) {
  unsigned b = (u & 0x80000000u) ? (u ^ 0x80000000u) : ~u;
  return __uint_as_float(b);
}

// ---------------------------------------------------------------------------
// Kernel 0: zero accumulators  acc[0]=sum d1(B), acc[1]=sum d2(B),
//                              acc[2]=sum d1(cd1), acc[3]=sum d2(cd1)
// ---------------------------------------------------------------------------
__global__ void zero_acc_kernel(float* __restrict__ acc) {
  if (threadIdx.x < 8) acc[threadIdx.x] = 0.0f;
}

// ---------------------------------------------------------------------------
// Kernel 1: per (b,n) — distances to all gt, rank-201 selection in LDS,
// write 200-neighbor list (padded to 208) + accumulate cd1.d1 term.
// One workgroup per (b,n); 8192 ordered-uint keys live in 32KB LDS.
// ---------------------------------------------------------------------------
__global__ __launch_bounds__(256) void select_knn_kernel(
    const float* __restrict__ means, const float* __restrict__ gt,
    float* __restrict__ grouped, float* __restrict__ acc) {
  __shared__ unsigned ordv[MM];      // 32 KB
  __shared__ unsigned redu[256];
  __shared__ int cnt;
  __shared__ int wcount;
  __shared__ int minTaken;

  const int tid = threadIdx.x;
  const int bn  = blockIdx.x;        // b*NN + n
  const int b   = bn / NN;

  const float* mp = means + (size_t)bn * 3;
  const float mx = mp[0], my = mp[1], mz = mp[2];

  // Pass 1: all distances -> ordered uints in LDS, track local min
  unsigned lmin = 0xFFFFFFFFu;
  for (int j = tid; j < MM; j += 256) {
    const float* g = gt + ((size_t)b * MM + j) * 3;
    float dx = mx - g[0], dy = my - g[1], dz = mz - g[2];
    float d  = dx * dx + dy * dy + dz * dz;
    unsigned u = f2ord(d);
    ordv[j] = u;
    lmin = (u < lmin) ? u : lmin;
  }
  redu[tid] = lmin;
  __syncthreads();
  for (int s = 128; s > 0; s >>= 1) {
    if (tid < s) redu[tid] = (redu[tid + s] < redu[tid]) ? redu[tid + s] : redu[tid];
    __syncthreads();
  }
  const unsigned minOrd = redu[0];
  __syncthreads();

  // Binary search for the (KK+1)-th smallest key: min v with count(<=v) >= 201
  unsigned lo = 0u, hi = 0xFFFFFFFFu;
  for (int it = 0; it < 32; ++it) {
    unsigned mid = lo + ((hi - lo) >> 1);
    if (tid == 0) cnt = 0;
    __syncthreads();
    int c = 0;
    for (int j = tid; j < MM; j += 256) c += (ordv[j] <= mid) ? 1 : 0;
    atomicAdd(&cnt, c);
    __syncthreads();
    const int total = cnt;
    if (total >= (KK + 1)) hi = mid; else lo = mid + 1u;
    __syncthreads();
  }
  const unsigned thr = hi;

  // Emit neighbor list: all keys <= thr minus exactly one copy of the min.
  if (tid == 0) { wcount = 0; minTaken = 0; }
  float* gout = grouped + (size_t)bn * KP * 3;
  for (int k = tid; k < KP; k += 256) {     // pad slots with far-away points
    gout[k * 3 + 0] = 1e18f; gout[k * 3 + 1] = 1e18f; gout[k * 3 + 2] = 1e18f;
  }
  __syncthreads();
  for (int j = tid; j < MM; j += 256) {
    unsigned u = ordv[j];
    if (u <= thr) {
      bool skip = false;
      if (u == minOrd) skip = (atomicAdd(&minTaken, 1) == 0);  // drop nearest once
      if (!skip) {
        int pos = atomicAdd(&wcount, 1);
        if (pos < KK) {
          const float* g = gt + ((size_t)b * MM + j) * 3;
          gout[pos * 3 + 0] = g[0];
          gout[pos * 3 + 1] = g[1];
          gout[pos * 3 + 2] = g[2];
        }
      }
    }
  }
  if (tid == 0) {
    float dmin = ord2f(minOrd);
    atomicAdd(&acc[2], sqrtf(fmaxf(dmin, EPSF)));   // cd1: mean over (b,n) later
  }
}

// ---------------------------------------------------------------------------
// Kernel 2: cd1.d2 — per (b,m) min over the 256 means (means cached in LDS).
// ---------------------------------------------------------------------------
__global__ __launch_bounds__(256) void gt_min_kernel(
    const float* __restrict__ means, const float* __restrict__ gt,
    float* __restrict__ acc) {
  __shared__ float sm[NN * 3];
  __shared__ float redf[256];
  const int tid = threadIdx.x;
  const int blocksPerB = MM / 256;                 // 32
  const int b = blockIdx.x / blocksPerB;
  const int m = (blockIdx.x % blocksPerB) * 256 + tid;

  for (int i = tid; i < NN * 3; i += 256) sm[i] = means[(size_t)b * NN * 3 + i];
  __syncthreads();

  const float* g = gt + ((size_t)b * MM + m) * 3;
  const float gx = g[0], gy = g[1], gz = g[2];
  float best = 3.4e38f;
  for (int n = 0; n < NN; ++n) {
    float dx = gx - sm[n * 3 + 0];
    float dy = gy - sm[n * 3 + 1];
    float dz = gz - sm[n * 3 + 2];
    best = fminf(best, dx * dx + dy * dy + dz * dz);
  }
  redf[tid] = sqrtf(fmaxf(best, EPSF));
  __syncthreads();
  for (int s = 128; s > 0; s >>= 1) {
    if (tid < s) redf[tid] += redf[tid + s];
    __syncthreads();
  }
  if (tid == 0) atomicAdd(&acc[3], redf[0]);
}

// ---------------------------------------------------------------------------
// Kernel 3: chamfer(sample_points[b,n], grouped[b,n]) with V_WMMA_F32_16X16X4
// Gram tiles. One workgroup (8 waves) per (b,n); wave w owns row tile w.
// d[s,k] = |a_s|^2 + |b_k|^2 - 2 a_s.b_k  (same formulation as reference).
// ---------------------------------------------------------------------------
__global__ __launch_bounds__(256) void chamfer_wmma_kernel(
    const float* __restrict__ sample_points, const float* __restrict__ grouped,
    float* __restrict__ acc) {
  __shared__ float sp4[SS][4];       // sample points, 4th coord = 0
  __shared__ float aas[SS];          // |a|^2
  __shared__ float gp4[KP][4];       // neighbors, padded
  __shared__ float bbs[KP];          // |b|^2 (pads are huge -> never win a min)
  __shared__ float colmin[8][KP];    // per-wave column minima
  __shared__ float redf[256];

  const int tid = threadIdx.x;
  const int bn  = blockIdx.x;

  for (int s = tid; s < SS; s += 256) {
    const float* p = sample_points + ((size_t)bn * SS + s) * 3;
    float x = p[0], y = p[1], z = p[2];
    sp4[s][0] = x; sp4[s][1] = y; sp4[s][2] = z; sp4[s][3] = 0.0f;
    aas[s] = x * x + y * y + z * z;
  }
  for (int k = tid; k < KP; k += 256) {
    const float* p = grouped + ((size_t)bn * KP + k) * 3;
    float x = p[0], y = p[1], z = p[2];
    gp4[k][0] = x; gp4[k][1] = y; gp4[k][2] = z; gp4[k][3] = 0.0f;
    bbs[k] = x * x + y * y + z * z;
  }
  __syncthreads();

  const int  wave = tid >> 5;        // 0..7, wave32
  const int  lane = tid & 31;
  const int  hl   = lane & 15;
  const bool up   = (lane >= 16);

  // A fragment (16x4 f32): lanes 0-15 -> (K0,K1), lanes 16-31 -> (K2,K3)
  const int arow = wave * 16 + hl;
  v2f afrag;
  afrag.x = up ? sp4[arow][2] : sp4[arow][0];
  afrag.y = up ? sp4[arow][3] : sp4[arow][1];

  // C/D layout: VGPR v -> M = v (lanes 0-15) or v+8 (lanes 16-31), N = hl
  const int rbase = wave * 16 + (up ? 8 : 0);
  float aav[8];
#pragma unroll
  for (int v = 0; v < 8; ++v) aav[v] = aas[rbase + v];

  float rowmin[8];
#pragma unroll
  for (int v = 0; v < 8; ++v) rowmin[v] = 3.4e38f;

  for (int t = 0; t < KP / 16; ++t) {          // 13 column tiles
    const int bcol = t * 16 + hl;
    v2f bfrag;                                  // B fragment (4x16 f32)
    bfrag.x = up ? gp4[bcol][2] : gp4[bcol][0];
    bfrag.y = up ? gp4[bcol][3] : gp4[bcol][1];

    v8f cfr = {0.f, 0.f, 0.f, 0.f, 0.f, 0.f, 0.f, 0.f};
    cfr = __builtin_amdgcn_wmma_f32_16x16x4_f32(
        /*neg_a=*/false, afrag, /*neg_b=*/false, bfrag,
        /*c_mod=*/(short)0, cfr, /*reuse_a=*/false, /*reuse_b=*/false);

    const float bbv = bbs[t * 16 + hl];
    float cmin = 3.4e38f;
#pragma unroll
    for (int v = 0; v < 8; ++v) {
      float d = aav[v] + bbv - 2.0f * cfr[v];
      rowmin[v] = fminf(rowmin[v], d);
      cmin = fminf(cmin, d);
    }
    // combine the two lane halves sharing column (t*16+hl)
    cmin = fminf(cmin, __shfl_xor(cmin, 16, 32));
    if (!up) colmin[wave][t * 16 + hl] = cmin;
  }

  // Row mins: butterfly within each 16-lane half (never crosses bit 4)
#pragma unroll
  for (int off = 1; off < 16; off <<= 1) {
#pragma unroll
    for (int v = 0; v < 8; ++v)
      rowmin[v] = fminf(rowmin[v], __shfl_xor(rowmin[v], off, 32));
  }
  float myd1 = 0.0f;
  if (hl == 0) {                      // lane 0 -> rows v, lane 16 -> rows v+8
#pragma unroll
    for (int v = 0; v < 8; ++v) myd1 += sqrtf(fmaxf(rowmin[v], EPSF));
  }
  redf[tid] = myd1;
  __syncthreads();
  for (int s = 128; s > 0; s >>= 1) {
    if (tid < s) redf[tid] += redf[tid + s];
    __syncthreads();
  }
  if (tid == 0) atomicAdd(&acc[0], redf[0]);
  __syncthreads();

  // Column mins across waves (only real K < 200)
  float myd2 = 0.0f;
  for (int k = tid; k < KK; k += 256) {
    float mval = colmin[0][k];
#pragma unroll
    for (int w = 1; w < 8; ++w) mval = fminf(mval, colmin[w][k]);
    myd2 += sqrtf(fmaxf(mval, EPSF));
  }
  redf[tid] = myd2;
  __syncthreads();
  for (int s = 128; s > 0; s >>= 1) {
    if (tid < s) redf[tid] += redf[tid + s];
    __syncthreads();
  }
  if (tid == 0) atomicAdd(&acc[1], redf[0]);
}

// ---------------------------------------------------------------------------
// Kernel 4: combine the 4 global sums into the 3 outputs.
// ---------------------------------------------------------------------------
__global__ void finalize_kernel(const float* __restrict__ acc,
                                float* __restrict__ out) {
  if (threadIdx.x == 0 && blockIdx.x == 0) {
    float cd2 = 0.5f * (acc[0] / (float)(BB * SS) + acc[1] / (float)(BB * KK));
    float cd1 = 0.5f * (acc[2] / (float)(BB * NN) + acc[3] / (float)(BB * MM));
    out[0] = cd2 * 1000.0f;
    out[1] = cd1 * 1000.0f;
    out[2] = cd2 * 1000.0f;
  }
}

extern "C" void kernel_launch(void* const* d_in, const int* in_sizes, int n_in,
                              void* d_out, int out_size, void* d_ws, size_t ws_size,
                              hipStream_t stream) {
  const float* means         = (const float*)d_in[0];  // (B,N,3)
  const float* sample_points = (const float*)d_in[1];  // (B,N,S,3)
  const float* gt            = (const float*)d_in[2];  // (B,M,3)
  float* out = (float*)d_out;                          // 3 floats

  // workspace: [0..63] accumulators, then grouped neighbors B*N*KP*3 floats
  float* acc     = (float*)d_ws;
  float* grouped = (float*)d_ws + 64;   // ~5.1 MB total

  hipLaunchKernelGGL(zero_acc_kernel, dim3(1), dim3(32), 0, stream, acc);
  hipLaunchKernelGGL(select_knn_kernel, dim3(BB * NN), dim3(256), 0, stream,
                     means, gt, grouped, acc);
  hipLaunchKernelGGL(gt_min_kernel, dim3(BB * (MM / 256)), dim3(256), 0, stream,
                     means, gt, acc);
  hipLaunchKernelGGL(chamfer_wmma_kernel, dim3(BB * NN), dim3(256), 0, stream,
                     sample_points, grouped, acc);
  hipLaunchKernelGGL(finalize_kernel, dim3(1), dim3(32), 0, stream, acc, out);
}